// KnnEdges_20968030339127
// MI455X (gfx1250) — compile-verified
//
#include <hip/hip_runtime.h>

// ---------------------------------------------------------------------------
// Batched brute-force kNN (B=4, M=8192, D=2, k=16) for MI455X / gfx1250.
// Gram-matrix tiles via V_WMMA_F32_16X16X4_F32 (wave32), LDS transpose of the
// 16x16 C tile, branch-free min/max bubble top-16 selection, atomic global
// max for the distance normalization.
// ---------------------------------------------------------------------------

typedef __attribute__((ext_vector_type(2))) float v2f;
typedef __attribute__((ext_vector_type(8))) float v8f;

#define BGRAPH 4
#define MNODES 8192
#define KNN    16
#define NTOT   (BGRAPH * MNODES)

// ---------------------------------------------------------------------------
// Gather pooled positions pos_p = pos[perm] and zero the global-max cell.
// ---------------------------------------------------------------------------
__global__ __launch_bounds__(256) void gather_pos_kernel(
    const float* __restrict__ pos, const int* __restrict__ perm,
    float* __restrict__ pos_out, unsigned int* __restrict__ maxcell, int n) {
  int i = blockIdx.x * blockDim.x + threadIdx.x;
  if (i == 0) *maxcell = 0u;
  if (i < n) {
    int p = perm[i];
    float2 v = *(const float2*)(pos + 2 * (size_t)p);
    *(float2*)(pos_out + 2 * (size_t)i) = v;
  }
}

// ---------------------------------------------------------------------------
// Passthrough copies.
// ---------------------------------------------------------------------------
__global__ __launch_bounds__(256) void copy4_kernel(
    const float4* __restrict__ src, float4* __restrict__ dst, int n4) {
  int i = blockIdx.x * blockDim.x + threadIdx.x;
  if (i < n4) dst[i] = src[i];
}

__global__ __launch_bounds__(256) void copyf_kernel(
    const float* __restrict__ src, float* __restrict__ dst, int n) {
  int i = blockIdx.x * blockDim.x + threadIdx.x;
  if (i < n) dst[i] = src[i];
}

__global__ __launch_bounds__(256) void i2f_kernel(
    const int* __restrict__ src, float* __restrict__ dst, int n) {
  int i = blockIdx.x * blockDim.x + threadIdx.x;
  if (i < n) dst[i] = (float)src[i];
}

// ---------------------------------------------------------------------------
// kNN kernel: one wave per 16-row tile; 8 waves (8 tiles) per block.
// ---------------------------------------------------------------------------
__global__ __launch_bounds__(256) void knn_wmma_kernel(
    const float* __restrict__ pos_p,      // [N,2] pooled positions
    float* __restrict__ ei_src,           // N*K (neighbor/source ids, as float)
    float* __restrict__ ei_tgt,           // N*K (target ids, as float)
    float* __restrict__ dist_ws,          // N*K unnormalized distances
    unsigned int* __restrict__ maxcell) { // global max(dist) as uint bits
  __shared__ float tile[8][16][17];       // +1 pad: no LDS bank conflicts
  __shared__ float rown[8][16];

  const int lane = threadIdx.x & 31;
  const int wave = threadIdx.x >> 5;
  const int l15  = lane & 15;
  const int hi   = lane >> 4;                     // 0: lanes 0-15, 1: 16-31
  const int tileId  = blockIdx.x * 8 + wave;      // 0 .. B*M/16-1
  const int g       = tileId >> 9;                // tileId / (M/16)
  const int rt      = tileId & 511;
  const int rowbase = g * MNODES + rt * 16;

  // --- A tile: 16x4 f32 (K=2..3 zero padding lives in lanes 16-31). ---
  // Unconditional load (lanes 16-31 reload lane&15's row) keeps EXEC uniform.
  const float2 pr = *(const float2*)(pos_p + 2 * (size_t)(rowbase + l15));
  rown[wave][l15] = pr.x * pr.x + pr.y * pr.y;    // duplicate writes, same val
  v2f a;
  a.x = hi ? 0.0f : pr.x;
  a.y = hi ? 0.0f : pr.y;
  __syncthreads();

  float rn[8];                                    // row norms this lane needs
#pragma unroll
  for (int i = 0; i < 8; ++i) rn[i] = rown[wave][i + 8 * hi];

  // --- Register-resident sorted top-16 (ascending d2). ---
  float topd[KNN];
  int   topi[KNN];
#pragma unroll
  for (int j = 0; j < KNN; ++j) { topd[j] = 3.0e38f; topi[j] = 0; }

  const float* cb = pos_p + 2 * (size_t)g * MNODES;

  for (int ct = 0; ct < MNODES / 16; ++ct) {
    // B tile: 4x16 f32, column l15 of this tile (L2-resident: 64KB/graph).
    const float2 pc = *(const float2*)(cb + 2 * (size_t)(ct * 16 + l15));
    if (ct + 1 < MNODES / 16) {
      __builtin_prefetch(cb + 2 * (size_t)((ct + 1) * 16 + l15), 0, 0);
    }
    const float cn = pc.x * pc.x + pc.y * pc.y;   // |q|^2 for col = lane&15
    v2f b;
    b.x = hi ? 0.0f : pc.x;
    b.y = hi ? 0.0f : pc.y;

    // D = A x B : 16x16 tile of dot products p.q (EXEC all-ones here).
    v8f c = {};
    c = __builtin_amdgcn_wmma_f32_16x16x4_f32(
        /*neg_a=*/false, a, /*neg_b=*/false, b,
        /*c_mod=*/(short)0, c, /*reuse_a=*/false, /*reuse_b=*/false);

    // C layout: VGPR i, lane l -> row (i + 8*(l>>4)), col (l&15).
#pragma unroll
    for (int i = 0; i < 8; ++i) {
      tile[wave][i + 8 * hi][l15] = rn[i] + cn - 2.0f * c[i];
    }
    __syncthreads();

    // 16 threads/wave each own one row: merge 16 candidates into top-16.
    if (lane < 16) {
#pragma unroll 1                                  // keep code compact
      for (int n = 0; n < 16; ++n) {
        float curd = tile[wave][lane][n];
        if (curd < topd[KNN - 1]) {
          int curi = ct * 16 + n;
          // Branch-free bubble insertion: per slot one v_cmp, v_min_num,
          // v_max_num and two v_cndmask; arrays stay in VGPRs (static idx).
#pragma unroll
          for (int j = 0; j < KNN; ++j) {
            const float od = topd[j];
            const int   oi = topi[j];
            const bool  sw = curd < od;
            topd[j] = fminf(curd, od);
            curd    = fmaxf(curd, od);
            topi[j] = sw ? curi : oi;
            curi    = sw ? oi : curi;
          }
        }
      }
    }
    __syncthreads();
  }

  // --- Emit edges + unnormalized distances; fold per-row max into atomic. ---
  if (lane < 16) {
    const int grow = rowbase + lane;
    float lmax = 0.0f;
#pragma unroll
    for (int j = 0; j < KNN; ++j) {
      const float d2v  = fmaxf(topd[j], 0.0f);
      const float dist = sqrtf(d2v);
      const size_t e = (size_t)grow * KNN + j;
      ei_src[e]  = (float)(g * MNODES + topi[j]);
      ei_tgt[e]  = (float)grow;
      dist_ws[e] = dist;
      lmax = fmaxf(lmax, dist);
    }
    atomicMax(maxcell, __float_as_uint(lmax));  // dist >= 0: bit-order == fp-order
  }
}

// ---------------------------------------------------------------------------
// edge_weight = dist / max(dist)
// ---------------------------------------------------------------------------
__global__ __launch_bounds__(256) void normalize_kernel(
    const float* __restrict__ dist_ws, const unsigned int* __restrict__ maxcell,
    float* __restrict__ ew, int n) {
  int i = blockIdx.x * blockDim.x + threadIdx.x;
  if (i < n) {
    const float mx = __uint_as_float(*maxcell);
    ew[i] = dist_ws[i] / mx;
  }
}

// ---------------------------------------------------------------------------
// Host launcher.
// Inputs: 0:x[N,64] f32  1:pos[N0,2] f32  2:edge_index  3:edge_weight
//         4:batch[N] int 5:perm[N] int    6:score[N] f32 7:i
// Output tuple (flattened): x, pos_p, edge_index[2,N*k], edge_weight[N*k],
//         batch, perm, score
// ---------------------------------------------------------------------------
extern "C" void kernel_launch(void* const* d_in, const int* in_sizes, int n_in,
                              void* d_out, int out_size, void* d_ws,
                              size_t ws_size, hipStream_t stream) {
  const float* x     = (const float*)d_in[0];
  const float* pos   = (const float*)d_in[1];
  const int*   batch = (const int*)d_in[4];
  const int*   perm  = (const int*)d_in[5];
  const float* score = (const float*)d_in[6];

  const int N  = in_sizes[4];     // 32768
  const int NK = N * KNN;

  float* out       = (float*)d_out;
  float* out_x     = out;                          // N*64
  float* out_pos   = out_x + (size_t)N * 64;       // N*2
  float* out_ei    = out_pos + (size_t)2 * N;      // 2*N*K (src then tgt)
  float* out_ew    = out_ei + (size_t)2 * NK;      // N*K
  float* out_batch = out_ew + (size_t)NK;          // N
  float* out_perm  = out_batch + (size_t)N;        // N
  float* out_score = out_perm + (size_t)N;         // N

  float*        ws_dist = (float*)d_ws;                              // N*K
  unsigned int* maxcell = (unsigned int*)((char*)d_ws + (size_t)NK * 4);

  const int T = 256;

  // 1) pos_p = pos[perm]; init max cell.
  gather_pos_kernel<<<(N + T - 1) / T, T, 0, stream>>>(pos, perm, out_pos,
                                                       maxcell, N);
  // 2) passthrough copies.
  const int n4 = (N * 64) / 4;
  copy4_kernel<<<(n4 + T - 1) / T, T, 0, stream>>>((const float4*)x,
                                                   (float4*)out_x, n4);
  i2f_kernel<<<(N + T - 1) / T, T, 0, stream>>>(batch, out_batch, N);
  i2f_kernel<<<(N + T - 1) / T, T, 0, stream>>>(perm, out_perm, N);
  copyf_kernel<<<(N + T - 1) / T, T, 0, stream>>>(score, out_score, N);

  // 3) kNN via WMMA Gram tiles: B*M/16 row tiles, 8 waves per block.
  const int nTiles = BGRAPH * (MNODES / 16);       // 2048
  knn_wmma_kernel<<<nTiles / 8, T, 0, stream>>>(out_pos, out_ei, out_ei + NK,
                                                ws_dist, maxcell);

  // 4) edge_weight = dist / global max.
  normalize_kernel<<<(NK + T - 1) / T, T, 0, stream>>>(ws_dist, maxcell,
                                                       out_ew, NK);
}